// RegionPartitioner_39204461478391
// MI455X (gfx1250) — compile-verified
//
#include <hip/hip_runtime.h>
#include <hip/hip_bf16.h>

// Region partitioner: x (8,4,500,500) f32 -> out (8,225,4,64,64) f32.
// Pure data movement; roofline ~150 MB HBM traffic (~6.4 us at 23.3 TB/s).
// Strategy: 3-D grid (tj, ti, b*4+c) so tile coords need no div/mod SALU;
// b128 coalesced loads (interior fast path), clamped scalar loads on the
// 12-col right edge, non-temporal b128 stores (output is write-once streaming,
// keep the 32 MB input resident in the 192 MB L2 for its ~3.7x reuse).

typedef float v4f __attribute__((ext_vector_type(4)));

#define DIM_H 500
#define DIM_W 500
#define N_C 4
#define N_B 8
#define RSZ 64                // region_size
#define STEP 32               // step
#define NTILE 15              // regions per spatial dim: (512-64)/32 + 1
#define NREG (NTILE * NTILE)  // 225

__global__ __launch_bounds__(256)
void RegionPartitioner_39204461478391_kernel(const float* __restrict__ x,
                                             float* __restrict__ out)
{
    // grid: x = tj (15), y = ti (15), z = b*N_C + c (32)
    const int tj = blockIdx.x;
    const int ti = blockIdx.y;
    const int z  = blockIdx.z;           // b*4 + c
    const int c  = z & (N_C - 1);
    const int b  = z >> 2;

    const float* __restrict__ src =
        x + (size_t)z * (size_t)(DIM_H * DIM_W);   // (b*4+c) plane

    // out flat block = ((b*NREG + ti*NTILE + tj) * N_C + c): multiplies only
    const int r = ti * NTILE + tj;
    float* __restrict__ dst =
        out + ((size_t)(b * NREG + r) * N_C + c) * (size_t)(RSZ * RSZ);

    const int t    = threadIdx.x;
    const int row0 = ti * STEP;
    const int col0 = tj * STEP;

    // Speculative L2 warm of the next region column (gfx1250 global_prefetch_b8)
    if (t == 0 && tj + 1 < NTILE) {
        __builtin_prefetch(src + (size_t)row0 * DIM_W + (col0 + STEP), 0, 1);
    }

    // 1024 float4 slots per 64x64 tile; 256 threads x 4 slots
    #pragma unroll
    for (int k = 0; k < 4; ++k) {
        const int f  = t + k * 256;     // float4 slot in [0,1024)
        const int i  = f >> 4;          // row within region [0,64)
        const int j0 = (f & 15) << 2;   // col within region, multiple of 4

        const int row = min(row0 + i, DIM_H - 1);   // edge-pad == clamp
        const int cb  = col0 + j0;
        const float* __restrict__ rp = src + (size_t)row * DIM_W;

        v4f v;
        if (cb + 3 < DIM_W) {
            // 16B-aligned: (row*500 + cb)*4 bytes is a multiple of 16
            v = *reinterpret_cast<const v4f*>(rp + cb);
        } else {
            // only tj==14, j0 in {52,56,60}: clamp columns to 499
            v.x = rp[min(cb + 0, DIM_W - 1)];
            v.y = rp[min(cb + 1, DIM_W - 1)];
            v.z = rp[min(cb + 2, DIM_W - 1)];
            v.w = rp[min(cb + 3, DIM_W - 1)];
        }
        // Write-once streaming output: non-temporal b128 store
        __builtin_nontemporal_store(v, reinterpret_cast<v4f*>(dst + f * 4));
    }
}

extern "C" void kernel_launch(void* const* d_in, const int* in_sizes, int n_in,
                              void* d_out, int out_size, void* d_ws, size_t ws_size,
                              hipStream_t stream) {
    (void)in_sizes; (void)n_in; (void)d_ws; (void)ws_size; (void)out_size;
    const float* x = (const float*)d_in[0];
    // d_in[1] = step (32), d_in[2] = region_size (64): fixed by the harness
    // shapes; device scalars cannot drive launch config under graph capture,
    // so they are baked into the kernel constants above.
    float* out = (float*)d_out;

    dim3 grid(NTILE, NTILE, N_B * N_C);   // 15 x 15 x 32 = 7200 tiles
    RegionPartitioner_39204461478391_kernel<<<grid, 256, 0, stream>>>(x, out);
}